// BiDirectionalGraphSAGE_16045997817885
// MI455X (gfx1250) — compile-verified
//
#include <hip/hip_runtime.h>
#include <hip/hip_bf16.h>

// ---------------------------------------------------------------------------
// BiDirectionalGraphSAGE for MI455X (gfx1250, wave32)
//   - edge scatter/mean: float4 gathers + global_atomic_add_f32 (L2-resident)
//   - SAGE linear step:  V_WMMA_F32_16X16X4_F32, one wave per 16x16 tile
//   - pooling: sorted-batch binary search, no atomics
// ---------------------------------------------------------------------------

typedef __attribute__((ext_vector_type(2))) float v2f;
typedef __attribute__((ext_vector_type(8))) float v8f;

#define HF 128                    // hidden/feature width
#define INV_BN 0.9999950000374997f   // 1/sqrt(1 + 1e-5)

// ---------------- per-direction in-degree counts ---------------------------
__global__ void count_kernel(const int* __restrict__ dst,
                             const int* __restrict__ edir,
                             float* __restrict__ cntP,
                             float* __restrict__ cntS, int E) {
    int e = blockIdx.x * blockDim.x + threadIdx.x;
    if (e >= E) return;
    int d = dst[e];
    if (edir[e] == 0) atomicAdd(cntP + d, 1.0f);
    else              atomicAdd(cntS + d, 1.0f);
}

// ---------------- masked edge scatter-add: agg[dst] += h[src] --------------
// one wave32 per edge; lane handles 4 consecutive features (float4 gather)
__global__ void scatter_kernel(const float* __restrict__ h,
                               const int* __restrict__ src,
                               const int* __restrict__ dst,
                               const int* __restrict__ edir, int dirval,
                               float* __restrict__ agg, int E) {
    int t = blockIdx.x * blockDim.x + threadIdx.x;
    int e = t >> 5;
    if (e >= E) return;                 // uniform per wave
    if (edir[e] != dirval) return;      // uniform per wave
    int lane = t & 31;
    int s = src[e], d = dst[e];
    const float4 v = *(const float4*)(h + (size_t)s * HF + lane * 4);
    float* o = agg + (size_t)d * HF + lane * 4;
    atomicAdd(o + 0, v.x);
    atomicAdd(o + 1, v.y);
    atomicAdd(o + 2, v.z);
    atomicAdd(o + 3, v.w);
}

// ---------------- agg -> mean (divide by max(cnt,1)) -----------------------
__global__ void mean_kernel(float* __restrict__ agg,
                            const float* __restrict__ cnt, int N) {
    int t = blockIdx.x * blockDim.x + threadIdx.x;
    int n = t >> 5;
    if (n >= N) return;
    int lane = t & 31;
    float c = cnt[n];
    if (c < 1.0f) c = 1.0f;
    float inv = 1.0f / c;
    float4* p = (float4*)(agg + (size_t)n * HF + lane * 4);
    float4 v = *p;
    v.x *= inv; v.y *= inv; v.z *= inv; v.w *= inv;
    *p = v;
}

// ---------------- fused SAGE linear: out = relu(mean@Wl^T + h@Wr^T + bl) ---
// One wave per 16x16 output tile.  FP32 WMMA, K stepped by 4.
// A-frag layout (ISA 7.12.2, 32-bit A 16x4): lanes 0-15 row M=l hold K=0(v0),
// K=1(v1); lanes 16-31 row M=l hold K=2(v0), K=3(v1) -> float2 @ koff={0,2}.
// B-frag mirrors with rows K striped across the two lane halves; B[k][col] =
// W[col*128 + k] (W^T access = contiguous row of W) -> float2 loads too.
__global__ __launch_bounds__(256)
void sage_gemm(const float* __restrict__ meanBuf, const float* __restrict__ h,
               const float* __restrict__ Wl, const float* __restrict__ bl,
               const float* __restrict__ Wr, float* __restrict__ out,
               int N, int nRowTiles) {
    int wave = blockIdx.x * (blockDim.x >> 5) + (threadIdx.x >> 5);
    if (wave >= nRowTiles * (HF / 16)) return;   // wave-uniform guard
    int lane = threadIdx.x & 31;
    int rowTile = wave >> 3;          // HF/16 == 8 column tiles
    int colTile = wave & 7;
    int half = lane >> 4;
    int l    = lane & 15;
    int row  = rowTile * 16 + l;
    if (row >= N) row = N - 1;        // clamp loads; stores are masked below
    int col  = colTile * 16 + l;
    int koff = half * 2;

    v8f acc = {};

    const float* aPtr = meanBuf + (size_t)row * HF + koff;
    const float* bPtr = Wl      + (size_t)col * HF + koff;
#pragma unroll 8
    for (int k = 0; k < HF; k += 4) {
        v2f a = *(const v2f*)(aPtr + k);
        v2f b = *(const v2f*)(bPtr + k);
        acc = __builtin_amdgcn_wmma_f32_16x16x4_f32(
            false, a, false, b, (short)0, acc, false, false);
    }
    aPtr = h  + (size_t)row * HF + koff;
    bPtr = Wr + (size_t)col * HF + koff;
#pragma unroll 8
    for (int k = 0; k < HF; k += 4) {
        v2f a = *(const v2f*)(aPtr + k);
        v2f b = *(const v2f*)(bPtr + k);
        acc = __builtin_amdgcn_wmma_f32_16x16x4_f32(
            false, a, false, b, (short)0, acc, false, false);
    }

    // C/D layout: VGPR r, lanes 0-15 -> (M=r, N=l); lanes 16-31 -> (M=r+8, N=l)
    float bias = bl[col];
#pragma unroll
    for (int r = 0; r < 8; ++r) {
        int orow = rowTile * 16 + r + half * 8;
        if (orow < N) {
            float v = acc[r] + bias;
            out[(size_t)orow * HF + col] = v > 0.0f ? v : 0.0f;
        }
    }
}

// ---------------- per-graph mean pooling + norm_emb affine -----------------
__device__ __forceinline__ int lower_bound_i(const int* __restrict__ a, int n, int key) {
    int lo = 0, hi = n;
    while (lo < hi) {
        int mid = (lo + hi) >> 1;
        if (a[mid] < key) lo = mid + 1; else hi = mid;
    }
    return lo;
}

__global__ void pool_kernel(const float* __restrict__ preOut,
                            const float* __restrict__ sucOut,
                            const int* __restrict__ batch,
                            const float* __restrict__ gnorm,
                            const float* __restrict__ bnorm,
                            float* __restrict__ combined, int N) {
    int b = blockIdx.x;           // one block per graph
    int f = threadIdx.x;          // 0..255 (2H features)
    int start = lower_bound_i(batch, N, b);
    int end   = lower_bound_i(batch, N, b + 1);
    const float* base = (f < HF) ? (preOut + f) : (sucOut + (f - HF));
    float s = 0.0f;
    for (int n = start; n < end; ++n) s += base[(size_t)n * HF];
    float cnt = (float)(end - start);
    if (cnt < 1.0f) cnt = 1.0f;
    float m = s / cnt;
    combined[(size_t)b * 256 + f] = gnorm[f] * (m * INV_BN) + bnorm[f];
}

// ---------------- tiny MLP heads (one thread per graph) --------------------
__global__ void head_kernel(const float* __restrict__ combined,
                            const float* __restrict__ gf,
                            const float* __restrict__ fc1_W, const float* __restrict__ fc1_b,
                            const float* __restrict__ bn1_g, const float* __restrict__ bn1_b,
                            const float* __restrict__ fc2_W, const float* __restrict__ fc2_b,
                            const float* __restrict__ fc3a_W, const float* __restrict__ fc3a_b,
                            const float* __restrict__ fc3b_W, const float* __restrict__ fc3b_b,
                            const float* __restrict__ normc_g, const float* __restrict__ normc_b,
                            float* __restrict__ out1, float* __restrict__ out2,
                            float* __restrict__ out3, int Bn) {
    int b = blockIdx.x * blockDim.x + threadIdx.x;
    if (b >= Bn) return;
    const float* c  = combined + (size_t)b * 256;
    const float* g5 = gf + (size_t)b * 5;     // G == 5
    float lev = g5[4];
    float nx0 = g5[2], nx1 = g5[3], nx2 = g5[4];
    float wf1 = g5[2], wf2 = g5[1];

    // fc1 (16 x 257) -> relu -> bn1
    float a1[16];
#pragma unroll
    for (int j = 0; j < 16; ++j) a1[j] = fc1_b[j];
    for (int i = 0; i < 256; ++i) {
        float ci = c[i];
#pragma unroll
        for (int j = 0; j < 16; ++j) a1[j] += ci * fc1_W[j * 257 + i];
    }
#pragma unroll
    for (int j = 0; j < 16; ++j) {
        float s = a1[j] + lev * fc1_W[j * 257 + 256];
        s = s > 0.0f ? s : 0.0f;
        out1[b * 16 + j] = bn1_g[j] * (s * INV_BN) + bn1_b[j];
    }

    // fc2 (2 x 259) -> relu; argmax
    float o2[2];
#pragma unroll
    for (int j = 0; j < 2; ++j) {
        const float* w = fc2_W + j * 259;
        float s = fc2_b[j];
        for (int i = 0; i < 256; ++i) s += c[i] * w[i];
        s += nx0 * w[256] + nx1 * w[257] + nx2 * w[258];
        s = s > 0.0f ? s : 0.0f;
        o2[j] = s;
        out2[b * 2 + j] = s;
    }
    int pred = (o2[1] > o2[0]) ? 1 : 0;   // argmax: ties -> 0

    // x3 = normc(concat(combined, wf1, wf2, nx3)); fc3a (4x261), fc3b (5x261)
    float accA[4], accB[5];
#pragma unroll
    for (int j = 0; j < 4; ++j) accA[j] = fc3a_b[j];
#pragma unroll
    for (int j = 0; j < 5; ++j) accB[j] = fc3b_b[j];
    float extra[5] = {wf1, wf2, nx0, nx1, nx2};
    for (int i = 0; i < 261; ++i) {
        float v  = (i < 256) ? c[i] : extra[i - 256];
        float x3 = normc_g[i] * (v * INV_BN) + normc_b[i];
#pragma unroll
        for (int j = 0; j < 4; ++j) accA[j] += x3 * fc3a_W[j * 261 + i];
#pragma unroll
        for (int j = 0; j < 5; ++j) accB[j] += x3 * fc3b_W[j * 261 + i];
    }
#pragma unroll
    for (int j = 0; j < 4; ++j) out3[b * 9 + j]     = (pred == 0) ? accA[j] : 0.0f;
#pragma unroll
    for (int j = 0; j < 5; ++j) out3[b * 9 + 4 + j] = (pred == 1) ? accB[j] : 0.0f;
}

// ---------------------------------------------------------------------------
extern "C" void kernel_launch(void* const* d_in, const int* in_sizes, int n_in,
                              void* d_out, int out_size, void* d_ws, size_t ws_size,
                              hipStream_t stream) {
    const float* x     = (const float*)d_in[0];
    const int*   eidx  = (const int*)  d_in[1];   // (2, E) flat
    const int*   edir  = (const int*)  d_in[2];
    const int*   batch = (const int*)  d_in[3];
    const float* gf    = (const float*)d_in[4];
    // d_in[5] = task (unused by reference)

    const float* W[18];                           // pre/suc x 3 layers x (Wl,bl,Wr)
    for (int i = 0; i < 18; ++i) W[i] = (const float*)d_in[6 + i];

    const float* norm_emb_g = (const float*)d_in[24];
    const float* norm_emb_b = (const float*)d_in[25];
    const float* fc1_W  = (const float*)d_in[26];
    const float* fc1_b  = (const float*)d_in[27];
    const float* bn1_g  = (const float*)d_in[28];
    const float* bn1_b  = (const float*)d_in[29];
    const float* fc2_W  = (const float*)d_in[30];
    const float* fc2_b  = (const float*)d_in[31];
    const float* fc3a_W = (const float*)d_in[32];
    const float* fc3a_b = (const float*)d_in[33];
    const float* fc3b_W = (const float*)d_in[34];
    const float* fc3b_b = (const float*)d_in[35];
    const float* normc_g = (const float*)d_in[36];
    const float* normc_b = (const float*)d_in[37];

    const int E  = in_sizes[2];        // edge_direction has E elements
    const int N  = in_sizes[3];        // batch has N elements
    const int Bn = in_sizes[4] / 5;    // graph_feature is (B, 5)
    const int* src = eidx;
    const int* dst = eidx + E;

    // workspace layout (floats)
    size_t NH = (size_t)N * HF;
    float* bufA = (float*)d_ws;        // agg/mean (per layer)
    float* bufB = bufA + NH;           // pre: h1, h3(out)
    float* bufC = bufB + NH;           // pre: h2 ; suc: h1, h3(out)
    float* bufD = bufC + NH;           // suc: h2
    float* cntP = bufD + NH;           // N
    float* cntS = cntP + N;            // N
    float* combined = cntS + N;        // B * 256

    // per-direction in-degrees (reused by all 3 layers of that direction)
    (void)hipMemsetAsync(cntP, 0, 2 * (size_t)N * sizeof(float), stream);
    count_kernel<<<(E + 255) / 256, 256, 0, stream>>>(dst, edir, cntP, cntS, E);

    float* outs_pre[3] = {bufB, bufC, bufB};
    float* outs_suc[3] = {bufC, bufD, bufC};
    int nRowTiles = (N + 15) / 16;
    int gemmBlocks = (nRowTiles * (HF / 16) + 7) / 8;   // 8 waves per block
    int scatBlocks = (int)(((size_t)E * 32 + 255) / 256);
    int meanBlocks = (int)(((size_t)N * 32 + 255) / 256);

    for (int dir = 0; dir < 2; ++dir) {
        const float* hcur = x;
        const float* cnt  = (dir == 0) ? cntP : cntS;
        for (int l = 0; l < 3; ++l) {
            const float* Wl = W[dir * 9 + l * 3 + 0];
            const float* bl = W[dir * 9 + l * 3 + 1];
            const float* Wr = W[dir * 9 + l * 3 + 2];
            float* hout = (dir == 0) ? outs_pre[l] : outs_suc[l];

            (void)hipMemsetAsync(bufA, 0, NH * sizeof(float), stream);
            scatter_kernel<<<scatBlocks, 256, 0, stream>>>(hcur, src, dst, edir,
                                                           dir, bufA, E);
            mean_kernel<<<meanBlocks, 256, 0, stream>>>(bufA, cnt, N);
            sage_gemm<<<gemmBlocks, 256, 0, stream>>>(bufA, hcur, Wl, bl, Wr,
                                                      hout, N, nRowTiles);
            hcur = hout;
        }
    }

    // pooling + heads
    pool_kernel<<<Bn, 256, 0, stream>>>(bufB, bufC, batch,
                                        norm_emb_g, norm_emb_b, combined, N);

    float* out1 = (float*)d_out;
    float* out2 = out1 + (size_t)Bn * 16;
    float* out3 = out2 + (size_t)Bn * 2;
    head_kernel<<<(Bn + 127) / 128, 128, 0, stream>>>(
        combined, gf, fc1_W, fc1_b, bn1_g, bn1_b, fc2_W, fc2_b,
        fc3a_W, fc3a_b, fc3b_W, fc3b_b, normc_g, normc_b,
        out1, out2, out3, Bn);
}